// CrossAttention_56547539419662
// MI455X (gfx1250) — compile-verified
//
#include <hip/hip_runtime.h>
#include <stdint.h>

typedef __attribute__((ext_vector_type(2))) float v2f;
typedef __attribute__((ext_vector_type(8))) float v8f;

#define B_ 64
#define L_ 16384
#define D_ 64
#define K_ 256
#define NBIN 4096
#define SPLITS 8

__device__ __forceinline__ unsigned ordf(float f) {
  unsigned u = __float_as_uint(f);
  return (u & 0x80000000u) ? ~u : (u | 0x80000000u);
}
__device__ __forceinline__ float unordf(unsigned u) {
  unsigned v = (u & 0x80000000u) ? (u & 0x7fffffffu) : ~u;
  return __uint_as_float(v);
}

__global__ void zero_hist_kernel(unsigned* __restrict__ hist, int n) {
  int i = blockIdx.x * blockDim.x + threadIdx.x;
  if (i < n) hist[i] = 0u;
}

// Grid: B_*SPLITS blocks x 256 threads (8 waves). Each block handles 128
// 16-key tiles of one batch. WMMA f32 16x16x4: A = query replicated over M,
// B = 16 keys over N, accumulate K=64 in 16 steps.
__global__ void __launch_bounds__(256)
sim_kernel(const float* __restrict__ query, const float* __restrict__ keys,
           const float* __restrict__ gp, float* __restrict__ sims,
           unsigned* __restrict__ hist) {
  __shared__ float qs[D_];
  __shared__ unsigned lhist[NBIN];
  __shared__ float sh_qscale;

  const int tid = threadIdx.x;
  const int batch = blockIdx.x / SPLITS;
  const int split = blockIdx.x % SPLITS;
  const int wave = tid >> 5;
  const int lane = tid & 31;
  const int ncol = lane & 15;          // key within tile (N column)
  const int koff = (lane >> 4) * 2;    // K sub-offset for this lane half

  for (int i = tid; i < NBIN; i += 256) lhist[i] = 0u;
  if (tid < D_) qs[tid] = query[batch * D_ + tid];
  __syncthreads();
  if (tid == 0) {
    float ss = 0.f;
    for (int d = 0; d < D_; ++d) ss += qs[d] * qs[d];
    sh_qscale = gp[0] * rsqrtf(ss);
  }
  __syncthreads();
  const float qscale = sh_qscale;

  // A fragments: query replicated across all 16 rows (constant per block)
  v2f qa[16];
#pragma unroll
  for (int ci = 0; ci < 16; ++ci) {
    qa[ci].x = qs[4 * ci + koff];
    qa[ci].y = qs[4 * ci + koff + 1];
  }

  const float* kbase = keys + (size_t)batch * L_ * D_;
  float* sb = sims + (size_t)batch * L_;

  const int tile0 = split * (L_ / 16 / SPLITS);  // 128 tiles per split
  for (int t = tile0 + wave; t < tile0 + 128; t += 8) {
    const float* kp = kbase + (size_t)(t * 16 + ncol) * D_ + koff;
    v2f kb[16];
#pragma unroll
    for (int ci = 0; ci < 16; ++ci)
      kb[ci] = *(const v2f*)(kp + 4 * ci);

    float ss = 0.f;
#pragma unroll
    for (int ci = 0; ci < 16; ++ci)
      ss += kb[ci].x * kb[ci].x + kb[ci].y * kb[ci].y;

    v8f acc = {0.f, 0.f, 0.f, 0.f, 0.f, 0.f, 0.f, 0.f};
#pragma unroll
    for (int ci = 0; ci < 16; ++ci)
      acc = __builtin_amdgcn_wmma_f32_16x16x4_f32(
          false, qa[ci], false, kb[ci], (short)0, acc, false, false);

    // combine per-lane-half sum-of-squares for this key
    ss += __shfl_xor(ss, 16, 32);
    float sim = acc[0] * qscale * rsqrtf(ss);
    if (lane < 16) {
      sb[t * 16 + ncol] = sim;
      atomicAdd(&lhist[ordf(sim) >> 20], 1u);
    }
  }
  __syncthreads();
  unsigned* gh = hist + batch * NBIN;
  for (int i = tid; i < NBIN; i += 256) {
    unsigned c = lhist[i];
    if (c) atomicAdd(&gh[i], c);
  }
}

// One block (256 threads) per batch: histogram threshold -> exact-ish top-256
// (deterministic, index-ordered tie handling) -> bitonic sort desc -> softmax.
__global__ void __launch_bounds__(256)
topk_kernel(const float* __restrict__ values, const float* __restrict__ sims,
            const unsigned* __restrict__ hist, float* __restrict__ out_vals,
            float* __restrict__ out_w) {
  __shared__ unsigned long long cand[K_];
  __shared__ int scanbuf[256];
  __shared__ float red[256];
  __shared__ int sh_tbin, sh_above, sh_ctr;

  const int tid = threadIdx.x;
  const int batch = blockIdx.x;
  const float* sb = sims + (size_t)batch * L_;
  const unsigned* gh = hist + batch * NBIN;

  if (tid == 0) {
    int total = 0, tbin = 0, above = 0;
    for (int b = NBIN - 1; b >= 0; --b) {
      int c = (int)gh[b];
      if (total + c >= K_) { tbin = b; above = total; break; }
      total += c;
    }
    sh_tbin = tbin; sh_above = above; sh_ctr = 0;
  }
  __syncthreads();
  const unsigned tbin = (unsigned)sh_tbin;
  const int above = sh_above;

  // admit all strictly-above-threshold elements (< 256 by construction)
  for (int i = tid; i < L_; i += 256) {
    unsigned u = ordf(sb[i]);
    if ((u >> 20) > tbin) {
      int p = atomicAdd(&sh_ctr, 1);
      cand[p] = ((unsigned long long)u << 32) | (unsigned)(~i);
    }
  }
  __syncthreads();

  // admit threshold-bin elements deterministically in index order
  const int chunk = L_ / 256;
  const int base = tid * chunk;
  int cnt = 0;
  for (int j = 0; j < chunk; ++j)
    if ((ordf(sb[base + j]) >> 20) == tbin) ++cnt;
  scanbuf[tid] = cnt;
  __syncthreads();
  for (int off = 1; off < 256; off <<= 1) {
    int add = (tid >= off) ? scanbuf[tid - off] : 0;
    __syncthreads();
    scanbuf[tid] += add;
    __syncthreads();
  }
  int r = scanbuf[tid] - cnt;  // exclusive rank among tie elements
  const int Q = K_ - above;
  for (int j = 0; j < chunk; ++j) {
    unsigned u = ordf(sb[base + j]);
    if ((u >> 20) == tbin) {
      if (r < Q)
        cand[above + r] =
            ((unsigned long long)u << 32) | (unsigned)(~(base + j));
      ++r;
    }
  }
  __syncthreads();

  // bitonic sort, descending; equal sims -> smaller index first (via ~idx)
  for (unsigned ksz = 2; ksz <= K_; ksz <<= 1) {
    for (unsigned j = ksz >> 1; j > 0; j >>= 1) {
      unsigned i = (unsigned)tid;
      unsigned ixj = i ^ j;
      if (ixj > i) {
        unsigned long long a = cand[i], c2 = cand[ixj];
        bool desc = ((i & ksz) == 0);
        if ((a < c2) == desc) { cand[i] = c2; cand[ixj] = a; }
      }
      __syncthreads();
    }
  }

  float sim = unordf((unsigned)(cand[tid] >> 32));
  float mx = unordf((unsigned)(cand[0] >> 32));
  float e = __expf(sim - mx);
  red[tid] = e;
  __syncthreads();
  for (int s = 128; s > 0; s >>= 1) {
    if (tid < s) red[tid] += red[tid + s];
    __syncthreads();
  }
  float w = e / red[0];
  unsigned idx = ~(unsigned)(cand[tid] & 0xffffffffull);
  out_vals[batch * K_ + tid] = values[(size_t)batch * L_ + idx];
  out_w[batch * K_ + tid] = w;
}

extern "C" void kernel_launch(void* const* d_in, const int* in_sizes, int n_in,
                              void* d_out, int out_size, void* d_ws,
                              size_t ws_size, hipStream_t stream) {
  (void)in_sizes; (void)n_in; (void)out_size; (void)ws_size;
  const float* query  = (const float*)d_in[0];
  const float* keys   = (const float*)d_in[1];
  const float* values = (const float*)d_in[2];
  const float* gp     = (const float*)d_in[3];
  // d_in[4] = k (== 256), compile-time constant here.

  float* sims = (float*)d_ws;                                     // 4 MB
  unsigned* hist = (unsigned*)((char*)d_ws + sizeof(float) * B_ * L_);  // 1 MB
  float* out_vals = (float*)d_out;
  float* out_w = (float*)d_out + B_ * K_;

  const int nh = B_ * NBIN;
  zero_hist_kernel<<<(nh + 255) / 256, 256, 0, stream>>>(hist, nh);
  sim_kernel<<<B_ * SPLITS, 256, 0, stream>>>(query, keys, gp, sims, hist);
  topk_kernel<<<B_, 256, 0, stream>>>(values, sims, hist, out_vals, out_w);
}